// DataConsistency_6777458393612
// MI455X (gfx1250) — compile-verified
//
#include <hip/hip_runtime.h>
#include <math.h>

// ---------------------------------------------------------------------------
// CDNA5 (gfx1250) CG-SENSE data consistency.
// 2D centered FFTs implemented as complex GEMMs on V_WMMA_F32_16X16X4_F32,
// with LDS-staged macro-tiles (async global->LDS when available).
// ---------------------------------------------------------------------------

typedef float v2f __attribute__((ext_vector_type(2)));
typedef float v8f __attribute__((ext_vector_type(8)));
typedef int   v4i __attribute__((ext_vector_type(4)));

#define HH   320
#define WW   320
#define NPIX 102400               // 320*320
#define NB   8
#define NC   16
#define BHW  (NB * NPIX)          // 819200
#define BCHW (NB * NC * NPIX)     // 13107200
#define CG_ITERS 10

// GEMM macro-tiling
#define KB   32                   // K block
#define MT   32                   // macro tile M (2 waves)
#define NT   64                   // macro tile N (4 waves)
#define A_STR 36                  // LDS row stride for A[m][k]  (conflict-free, 16B rows)
#define B_STR 72                  // LDS row stride for B[k][n]  (conflict-free, 16B rows)

#if defined(__gfx1250__) && \
    __has_builtin(__builtin_amdgcn_global_load_async_to_lds_b128) && \
    __has_builtin(__builtin_amdgcn_s_wait_asynccnt)
#define USE_ASYNC_LDS 1
#else
#define USE_ASYNC_LDS 0
#endif

// workspace layout (in floats)
static constexpr size_t oFr  = 0;
static constexpr size_t oFi  = oFr  + NPIX;     // centered DFT matrix (fwd)
static constexpr size_t oR   = oFi  + NPIX;     // r  (re)
static constexpr size_t oRi  = oR   + BHW;      // r  (im)
static constexpr size_t oP   = oRi  + BHW;      // p  (re)
static constexpr size_t oPi  = oP   + BHW;      // p  (im)
static constexpr size_t oB   = oPi  + BHW;      // b  (re)
static constexpr size_t oBi  = oB   + BHW;      // b  (im)
static constexpr size_t oQ   = oBi  + BHW;      // q  (re)
static constexpr size_t oQi  = oQ   + BHW;      // q  (im)
static constexpr size_t oCP  = oQi  + BHW;      // coil buffer (re)
static constexpr size_t oCPi = oCP  + BCHW;     // coil buffer (im)
static constexpr size_t oT   = oCPi + BCHW;     // gemm temp (re)
static constexpr size_t oTi  = oT   + BCHW;     // gemm temp (im)
static constexpr size_t oS   = oTi  + BCHW;     // scalars[8]

// ---------------------------------------------------------------------------
// Centered ortho DFT matrix: F[k,n] = exp(-2*pi*i*(k-160)(n-160)/320)/sqrt(320)
// 2D centered FFT: Y = F * X * F (F symmetric); inverse = conj(F) both sides.
// ---------------------------------------------------------------------------
__global__ void init_dft_kernel(float* __restrict__ Fr, float* __restrict__ Fi) {
    int idx = blockIdx.x * blockDim.x + threadIdx.x;
    if (idx >= NPIX) return;
    int k = idx / WW, n = idx % WW;
    int ph = ((k - 160) * (n - 160)) % 320;       // integer phase mod N (exact)
    float theta = 6.28318530717958647692f * (float)ph * (1.0f / 320.0f);
    float s, c;
    sincosf(theta, &s, &c);
    const float inv = 0.05590169943749474f;       // 1/sqrt(320)
    Fr[idx] = c * inv;
    Fi[idx] = -s * inv;
}

// r = us + mu*recon ; p = r ; b = 0
__global__ void init_state_kernel(const float* __restrict__ us,
                                  const float* __restrict__ recon,
                                  const float* __restrict__ mu,
                                  float* __restrict__ rr, float* __restrict__ ri,
                                  float* __restrict__ pr, float* __restrict__ pi,
                                  float* __restrict__ br, float* __restrict__ bi) {
    int idx = blockIdx.x * blockDim.x + threadIdx.x;
    if (idx >= BHW) return;
    int b = idx / NPIX, pix = idx % NPIX;
    float m = mu[0];
    float ur = us[(size_t)(b * 2 + 0) * NPIX + pix];
    float ui = us[(size_t)(b * 2 + 1) * NPIX + pix];
    float vr = recon[(size_t)(b * 2 + 0) * NPIX + pix];
    float vi = recon[(size_t)(b * 2 + 1) * NPIX + pix];
    float xr = fmaf(m, vr, ur);
    float xi = fmaf(m, vi, ui);
    rr[idx] = xr; ri[idx] = xi;
    pr[idx] = xr; pi[idx] = xi;
    br[idx] = 0.0f; bi[idx] = 0.0f;
}

// coil projection: cp[b,c] = p[b] * csm[b,c]
__global__ void coil_proj_kernel(const float* __restrict__ pr, const float* __restrict__ pi,
                                 const float* __restrict__ sr, const float* __restrict__ si,
                                 float* __restrict__ cr, float* __restrict__ ci) {
    int idx = blockIdx.x * blockDim.x + threadIdx.x;
    if (idx >= BCHW) return;
    int pix = idx % NPIX;
    int bc  = idx / NPIX;
    int b   = bc >> 4;
    size_t pidx = (size_t)b * NPIX + pix;
    float xr = pr[pidx], xi = pi[pidx];
    float ar = sr[idx],  ai = si[idx];
    cr[idx] = xr * ar - xi * ai;
    ci[idx] = xr * ai + xi * ar;
}

// ---------------------------------------------------------------------------
// Batched complex GEMM on V_WMMA_F32_16X16X4_F32 with LDS macro-tile staging.
//   C = (Ar + i*sa*Ai) * (Br + i*sb*Bi)      (all matrices 320x320 row-major)
// Block: 256 thr = 8 waves (2x4), 32x64 macro tile, K blocked by 32.
// A fragment (16x4): lanes 0-15 M=lane K={0,1}; lanes 16-31 M=lane-16 K={2,3}.
// B fragment (4x16): lanes 0-15 N=lane K={0,1}; lanes 16-31 N=lane-16 K={2,3}.
// C/D (16x16): vgpr r, lanes 0-15 -> M=r,N=lane; lanes 16-31 -> M=r+8.
// LDS strides chosen bank-conflict-free for the wave32 K-split lane groups.
// ---------------------------------------------------------------------------
__global__ __launch_bounds__(256)
void cgemm_wmma_kernel(const float* __restrict__ Ar, const float* __restrict__ Ai,
                       float sa,
                       const float* __restrict__ Br, const float* __restrict__ Bi,
                       float sb,
                       float* __restrict__ Cr, float* __restrict__ Ci,
                       size_t strideA, size_t strideB, size_t strideC) {
    __shared__ float As_r[MT * A_STR];     // A[m][k], row stride 36
    __shared__ float As_i[MT * A_STR];
    __shared__ float Bs_r[KB * B_STR];     // B[k][n], row stride 72
    __shared__ float Bs_i[KB * B_STR];

    const int tid    = threadIdx.x;
    const int wave   = tid >> 5;
    const int lane   = tid & 31;
    const int l15    = lane & 15;
    const int kh     = lane >> 4;                 // 0: K pair {0,1}, 1: {2,3}
    const int wave_m = wave & 1;                  // 0..1
    const int wave_n = wave >> 1;                 // 0..3
    const int bm     = blockIdx.x / 5;            // 10 macro rows
    const int bn     = blockIdx.x % 5;            // 5 macro cols
    const int m0     = bm * MT;
    const int n0     = bn * NT;
    const size_t batch = blockIdx.y;

    const float* gAr = Ar + batch * strideA;
    const float* gAi = Ai + batch * strideA;
    const float* gBr = Br + batch * strideB;
    const float* gBi = Bi + batch * strideB;

    // staging decomposition
    const int a_row = tid >> 3;                   // 0..31 (m)
    const int a_q   = (tid & 7) << 2;             // k offset 0..28
    const int b_row = tid >> 4;                   // 0..15 (k, +16 second half)
    const int b_c4  = (tid & 15) << 2;            // n offset 0..60

    const int mloc = wave_m * 16 + l15;
    const int nloc = wave_n * 16 + l15;

    v8f accRR = {}, accII = {}, accRI = {}, accIR = {};

    for (int k0 = 0; k0 < WW; k0 += KB) {
        __syncthreads();                          // protect LDS from previous block
        {
            size_t ga = (size_t)(m0 + a_row) * WW + k0 + a_q;
            size_t la = (size_t)a_row * A_STR + a_q;
#if USE_ASYNC_LDS
            __builtin_amdgcn_global_load_async_to_lds_b128((v4i*)(gAr + ga), (v4i*)&As_r[la], 0, 0);
            __builtin_amdgcn_global_load_async_to_lds_b128((v4i*)(gAi + ga), (v4i*)&As_i[la], 0, 0);
#else
            *(float4*)&As_r[la] = *(const float4*)(gAr + ga);
            *(float4*)&As_i[la] = *(const float4*)(gAi + ga);
#endif
        }
#pragma unroll
        for (int h = 0; h < 2; ++h) {
            int kr = b_row + h * 16;
            size_t gb = (size_t)(k0 + kr) * WW + n0 + b_c4;
            size_t lb = (size_t)kr * B_STR + b_c4;
#if USE_ASYNC_LDS
            __builtin_amdgcn_global_load_async_to_lds_b128((v4i*)(gBr + gb), (v4i*)&Bs_r[lb], 0, 0);
            __builtin_amdgcn_global_load_async_to_lds_b128((v4i*)(gBi + gb), (v4i*)&Bs_i[lb], 0, 0);
#else
            *(float4*)&Bs_r[lb] = *(const float4*)(gBr + gb);
            *(float4*)&Bs_i[lb] = *(const float4*)(gBi + gb);
#endif
        }
#if USE_ASYNC_LDS
        __builtin_amdgcn_s_wait_asynccnt(0);
#endif
        __syncthreads();

#pragma unroll
        for (int kk = 0; kk < KB; kk += 4) {
            int kp = kk + 2 * kh;
            // A fragments: 8B-aligned ds_load_b64 (m*36 even, kp even)
            v2f fa_r = *reinterpret_cast<const v2f*>(&As_r[mloc * A_STR + kp]);
            v2f fa_i = *reinterpret_cast<const v2f*>(&As_i[mloc * A_STR + kp]);
            // B fragments: two rows, B_STR apart
            v2f fb_r = { Bs_r[kp * B_STR + nloc], Bs_r[(kp + 1) * B_STR + nloc] };
            v2f fb_i = { Bs_i[kp * B_STR + nloc], Bs_i[(kp + 1) * B_STR + nloc] };

            accRR = __builtin_amdgcn_wmma_f32_16x16x4_f32(false, fa_r, false, fb_r,
                                                          (short)0, accRR, false, false);
            accII = __builtin_amdgcn_wmma_f32_16x16x4_f32(false, fa_i, false, fb_i,
                                                          (short)0, accII, false, false);
            accRI = __builtin_amdgcn_wmma_f32_16x16x4_f32(false, fa_r, false, fb_i,
                                                          (short)0, accRI, false, false);
            accIR = __builtin_amdgcn_wmma_f32_16x16x4_f32(false, fa_i, false, fb_r,
                                                          (short)0, accIR, false, false);
        }
    }

    const float sab = sa * sb;
    float* cr = Cr + batch * strideC;
    float* ci = Ci + batch * strideC;
#pragma unroll
    for (int r = 0; r < 8; ++r) {
        int m = m0 + wave_m * 16 + r + 8 * kh;
        int n = n0 + wave_n * 16 + l15;
        size_t idx = (size_t)m * WW + n;
        cr[idx] = fmaf(-sab, accII[r], accRR[r]);
        ci[idx] = fmaf(sb, accRI[r], sa * accIR[r]);
    }
}

// kspace masking: cp[b,c,:,:] *= mask[b,:,:]
__global__ void mask_kernel(float* __restrict__ cr, float* __restrict__ ci,
                            const float* __restrict__ mask) {
    int idx = blockIdx.x * blockDim.x + threadIdx.x;
    if (idx >= BCHW) return;
    int pix = idx % NPIX;
    int b   = (idx / NPIX) >> 4;
    float m = mask[(size_t)b * NPIX + pix];
    cr[idx] *= m;
    ci[idx] *= m;
}

// q[b] = sum_c cp[b,c]*conj(csm[b,c]) + mu*p[b]
__global__ void coil_combine_kernel(const float* __restrict__ cr, const float* __restrict__ ci,
                                    const float* __restrict__ sr, const float* __restrict__ si,
                                    const float* __restrict__ pr, const float* __restrict__ pi,
                                    const float* __restrict__ mu,
                                    float* __restrict__ qr, float* __restrict__ qi) {
    int idx = blockIdx.x * blockDim.x + threadIdx.x;
    if (idx >= BHW) return;
    int b = idx / NPIX, pix = idx % NPIX;
    float accr = 0.0f, acci = 0.0f;
#pragma unroll
    for (int c = 0; c < NC; ++c) {
        size_t k = (size_t)(b * NC + c) * NPIX + pix;
        float xr = cr[k], xi = ci[k];
        float ar = sr[k], ai = si[k];
        accr = fmaf(xr, ar, fmaf(xi, ai, accr));    // re(x*conj(s))
        acci = fmaf(xi, ar, fmaf(-xr, ai, acci));   // im(x*conj(s))
    }
    float m = mu[0];
    qr[idx] = fmaf(m, pr[idx], accr);
    qi[idx] = fmaf(m, pi[idx], acci);
}

__global__ void zero_scalars_kernel(float* __restrict__ s) {
    if (threadIdx.x < 4) s[threadIdx.x] = 0.0f;   // rr, pq_re, pq_im, rn
}

// rr = sum |r|^2 ; pq = sum q*conj(p)   (block tree-reduce + global atomics)
__global__ void reduce_rr_pq_kernel(const float* __restrict__ rr_, const float* __restrict__ ri_,
                                    const float* __restrict__ pr, const float* __restrict__ pi,
                                    const float* __restrict__ qr, const float* __restrict__ qi,
                                    float* __restrict__ s) {
    __shared__ float s0[256], s1[256], s2[256];
    int idx = blockIdx.x * blockDim.x + threadIdx.x;   // exact: 3200*256 == BHW
    float xr = rr_[idx], xi = ri_[idx];
    float ar = pr[idx],  ai = pi[idx];
    float br = qr[idx],  bi = qi[idx];
    s0[threadIdx.x] = fmaf(xr, xr, xi * xi);
    s1[threadIdx.x] = fmaf(br, ar, bi * ai);           // re(q*conj(p))
    s2[threadIdx.x] = fmaf(bi, ar, -br * ai);          // im(q*conj(p))
    __syncthreads();
    for (int off = 128; off > 0; off >>= 1) {
        if (threadIdx.x < off) {
            s0[threadIdx.x] += s0[threadIdx.x + off];
            s1[threadIdx.x] += s1[threadIdx.x + off];
            s2[threadIdx.x] += s2[threadIdx.x + off];
        }
        __syncthreads();
    }
    if (threadIdx.x == 0) {
        atomicAdd(&s[0], s0[0]);
        atomicAdd(&s[1], s1[0]);
        atomicAdd(&s[2], s2[0]);
    }
}

// alpha = rr / pq  (complex)
__global__ void scalar_alpha_kernel(float* __restrict__ s) {
    float rr = s[0], pr = s[1], pi = s[2];
    float den = fmaf(pr, pr, pi * pi);
    s[4] = rr * pr / den;
    s[5] = -rr * pi / den;
}

// b += alpha*p ; r -= alpha*q ; rn += |r_new|^2
__global__ void update1_kernel(float* __restrict__ br, float* __restrict__ bi,
                               float* __restrict__ rr_, float* __restrict__ ri_,
                               const float* __restrict__ pr, const float* __restrict__ pi,
                               const float* __restrict__ qr, const float* __restrict__ qi,
                               float* __restrict__ s) {
    __shared__ float s0[256];
    int idx = blockIdx.x * blockDim.x + threadIdx.x;
    float ar = s[4], ai = s[5];
    float xr = pr[idx], xi = pi[idx];
    br[idx] += ar * xr - ai * xi;
    bi[idx] += ar * xi + ai * xr;
    float yr = qr[idx], yi = qi[idx];
    float nr = rr_[idx] - (ar * yr - ai * yi);
    float ni = ri_[idx] - (ar * yi + ai * yr);
    rr_[idx] = nr; ri_[idx] = ni;
    s0[threadIdx.x] = fmaf(nr, nr, ni * ni);
    __syncthreads();
    for (int off = 128; off > 0; off >>= 1) {
        if (threadIdx.x < off) s0[threadIdx.x] += s0[threadIdx.x + off];
        __syncthreads();
    }
    if (threadIdx.x == 0) atomicAdd(&s[3], s0[0]);
}

__global__ void scalar_beta_kernel(float* __restrict__ s) {
    s[6] = s[3] / s[0];
}

// p = r + beta*p
__global__ void update2_kernel(float* __restrict__ pr, float* __restrict__ pi,
                               const float* __restrict__ rr_, const float* __restrict__ ri_,
                               const float* __restrict__ s) {
    int idx = blockIdx.x * blockDim.x + threadIdx.x;
    float beta = s[6];
    pr[idx] = fmaf(beta, pr[idx], rr_[idx]);
    pi[idx] = fmaf(beta, pi[idx], ri_[idx]);
}

// out[b,0]=b_re, out[b,1]=b_im
__global__ void pack_kernel(const float* __restrict__ br, const float* __restrict__ bi,
                            float* __restrict__ out) {
    int idx = blockIdx.x * blockDim.x + threadIdx.x;
    if (idx >= BHW) return;
    int b = idx / NPIX, pix = idx % NPIX;
    out[(size_t)(b * 2 + 0) * NPIX + pix] = br[idx];
    out[(size_t)(b * 2 + 1) * NPIX + pix] = bi[idx];
}

// ---------------------------------------------------------------------------
extern "C" void kernel_launch(void* const* d_in, const int* in_sizes, int n_in,
                              void* d_out, int out_size, void* d_ws, size_t ws_size,
                              hipStream_t stream) {
    (void)in_sizes; (void)n_in; (void)out_size; (void)ws_size;
    const float* us    = (const float*)d_in[0];
    const float* recon = (const float*)d_in[1];
    const float* mask  = (const float*)d_in[2];
    const float* csm_r = (const float*)d_in[3];
    const float* csm_i = (const float*)d_in[4];
    const float* mu    = (const float*)d_in[5];
    float* out = (float*)d_out;
    float* w   = (float*)d_ws;

    float *Fr = w + oFr, *Fi = w + oFi;
    float *rr_ = w + oR,  *ri_ = w + oRi;
    float *pr  = w + oP,  *pi_ = w + oPi;
    float *br  = w + oB,  *bi_ = w + oBi;
    float *qr  = w + oQ,  *qi_ = w + oQi;
    float *cpr = w + oCP, *cpi = w + oCPi;
    float *tr  = w + oT,  *ti  = w + oTi;
    float *sc  = w + oS;

    const dim3 blk(256);
    const dim3 gPix(NPIX / 256);          // 400
    const dim3 gBHW(BHW / 256);           // 3200
    const dim3 gBCHW(BCHW / 256);         // 51200
    const dim3 gGemm(50, NB * NC);        // 10x5 macro tiles, 128 images

    init_dft_kernel<<<gPix, blk, 0, stream>>>(Fr, Fi);
    init_state_kernel<<<gBHW, blk, 0, stream>>>(us, recon, mu, rr_, ri_, pr, pi_, br, bi_);

    for (int it = 0; it < CG_ITERS; ++it) {
        zero_scalars_kernel<<<1, 32, 0, stream>>>(sc);

        // coil projection
        coil_proj_kernel<<<gBCHW, blk, 0, stream>>>(pr, pi_, csm_r, csm_i, cpr, cpi);

        // forward centered FFT: cp <- F * cp * F
        cgemm_wmma_kernel<<<gGemm, blk, 0, stream>>>(Fr, Fi, 1.0f, cpr, cpi, 1.0f,
                                                     tr, ti, 0, NPIX, NPIX);
        cgemm_wmma_kernel<<<gGemm, blk, 0, stream>>>(tr, ti, 1.0f, Fr, Fi, 1.0f,
                                                     cpr, cpi, NPIX, 0, NPIX);
        // k-space mask
        mask_kernel<<<gBCHW, blk, 0, stream>>>(cpr, cpi, mask);

        // inverse centered FFT: cp <- conj(F) * cp * conj(F)
        cgemm_wmma_kernel<<<gGemm, blk, 0, stream>>>(Fr, Fi, -1.0f, cpr, cpi, 1.0f,
                                                     tr, ti, 0, NPIX, NPIX);
        cgemm_wmma_kernel<<<gGemm, blk, 0, stream>>>(tr, ti, 1.0f, Fr, Fi, -1.0f,
                                                     cpr, cpi, NPIX, 0, NPIX);

        // coil combine + Tikhonov
        coil_combine_kernel<<<gBHW, blk, 0, stream>>>(cpr, cpi, csm_r, csm_i,
                                                      pr, pi_, mu, qr, qi_);

        // CG scalar updates
        reduce_rr_pq_kernel<<<gBHW, blk, 0, stream>>>(rr_, ri_, pr, pi_, qr, qi_, sc);
        scalar_alpha_kernel<<<1, 1, 0, stream>>>(sc);
        update1_kernel<<<gBHW, blk, 0, stream>>>(br, bi_, rr_, ri_, pr, pi_, qr, qi_, sc);
        scalar_beta_kernel<<<1, 1, 0, stream>>>(sc);
        update2_kernel<<<gBHW, blk, 0, stream>>>(pr, pi_, rr_, ri_, sc);
    }

    pack_kernel<<<gBHW, blk, 0, stream>>>(br, bi_, out);
}